// BicliqueAttentionLayer_37194416783611
// MI455X (gfx1250) — compile-verified
//
#include <hip/hip_runtime.h>
#include <math.h>

#define N_NODES 50000
#define N_EDGES 1600000
#define DIM 128
#define NEG_SLOPE 0.01f

typedef __attribute__((ext_vector_type(2))) float v2f;
typedef __attribute__((ext_vector_type(8))) float v8f;

// ---- float <-> monotonic-uint encoding for atomic float max ----
__device__ __forceinline__ unsigned fenc(float f) {
    unsigned u = __float_as_uint(f);
    return (u & 0x80000000u) ? ~u : (u | 0x80000000u);
}
__device__ __forceinline__ float fdec(unsigned u) {
    return (u & 0x80000000u) ? __uint_as_float(u ^ 0x80000000u) : __uint_as_float(~u);
}

// ---- kernel 0a: init per-node arrays (mx = enc(-inf), denom = 0) ----
__global__ void init_nodes_k(unsigned* __restrict__ mx_enc, float* __restrict__ denom) {
    int i = blockIdx.x * blockDim.x + threadIdx.x;
    if (i < N_NODES) {
        mx_enc[i] = 0x007FFFFFu;   // fenc(-inf)
        denom[i]  = 0.0f;
    }
}

// ---- kernel 0b: zero output accumulator ----
__global__ void zero_out_k(float* __restrict__ out) {
    size_t i = (size_t)blockIdx.x * blockDim.x + threadIdx.x;
    if (i < (size_t)N_NODES * DIM) out[i] = 0.0f;
}

// ---- kernel 0c: fold feature mask into W and pack K-pairs ----
// (feat * sigmoid(mask)) @ W == feat @ (diag(sigmoid(mask)) * W).
// Pack so that for even k, the pair (k, k+1) a lane needs sits in one b64:
//   Wp[((k>>1)*DIM + n)*2 + (k&1)] = sigmoid(mask[k]) * W[k*DIM + n]
__global__ void pack_w_k(const float* __restrict__ W,
                         const float* __restrict__ mask_w,
                         float* __restrict__ Wp) {
    int idx = blockIdx.x * blockDim.x + threadIdx.x;   // 0..16383
    if (idx >= DIM * DIM) return;
    const int k = idx >> 7;        // row of W (K index)
    const int n = idx & 127;       // column
    const float sig = 1.0f / (1.0f + expf(-mask_w[k]));
    Wp[(((size_t)(k >> 1) * DIM + n) << 1) + (k & 1)] = sig * W[idx];
}

// ---- kernel 1: h = feat @ Wp + b via fp32 WMMA 16x16x4 ----
// grid.x = 3125 row tiles (16 rows); block = (32, 8): one wave per 16x16 tile,
// threadIdx.y = column tile. Inner loop: 2x global_load_b64 + 1x v_wmma.
__global__ void proj_wmma_k(const float* __restrict__ feat,
                            const float* __restrict__ Wp,
                            const float* __restrict__ bias,
                            float* __restrict__ h) {
    const int lane  = threadIdx.x;           // 0..31
    const int row0  = blockIdx.x * 16;
    const int col0  = threadIdx.y * 16;
    const int m     = lane & 15;             // A row (lanes 0-15 == 16-31)
    const int n     = lane & 15;             // B/C column within tile
    const int khalf = (lane >> 4) << 1;      // 0 for lanes 0-15, 2 for 16-31

    const float* arow = feat + (size_t)(row0 + m) * DIM;
    const float* bcol = Wp + (size_t)(col0 + n) * 2;   // + (k>>1)*DIM*2 per step
    v8f c = {};
    #pragma unroll 8
    for (int kb = 0; kb < 32; ++kb) {        // K = 128 = 32 steps of K=4
        const int k = kb * 4 + khalf;        // even
        const v2f a = *(const v2f*)(arow + k);
        const v2f b = *(const v2f*)(bcol + (size_t)(k >> 1) * (DIM * 2));
        c = __builtin_amdgcn_wmma_f32_16x16x4_f32(false, a, false, b,
                                                  (short)0, c, false, false);
    }
    // C/D layout: VGPR r -> row r (lanes 0-15) / row r+8 (lanes 16-31)
    const int mhi = (lane >> 4) * 8;
    const float bv = bias[col0 + n];
    #pragma unroll
    for (int r = 0; r < 8; ++r)
        h[(size_t)(row0 + r + mhi) * DIM + col0 + n] = c[r] + bv;
}

// ---- kernel 2: per-node attention GEMVs: s_src = h.aw[:d], s_dst = h.aw[d:] ----
// one wave32 per node, shuffle reduction
__global__ void node_scores_k(const float* __restrict__ h,
                              const float* __restrict__ attn_w,
                              float* __restrict__ s_src,
                              float* __restrict__ s_dst) {
    const int lane = threadIdx.x & 31;
    const int node = (blockIdx.x * blockDim.x + threadIdx.x) >> 5;
    if (node >= N_NODES) return;
    const float* hr = h + (size_t)node * DIM;
    float a0 = 0.0f, a1 = 0.0f;
    #pragma unroll
    for (int k = lane; k < DIM; k += 32) {
        const float hv = hr[k];
        a0 += hv * attn_w[k];
        a1 += hv * attn_w[DIM + k];
    }
    #pragma unroll
    for (int off = 16; off > 0; off >>= 1) {
        a0 += __shfl_down(a0, off);
        a1 += __shfl_down(a1, off);
    }
    if (lane == 0) { s_src[node] = a0; s_dst[node] = a1; }
}

// ---- kernel 3: edge scores + leaky relu + segment max (atomic on encoded u32) ----
__global__ void edge_score_k(const int* __restrict__ src, const int* __restrict__ dst,
                             const float* __restrict__ s_src, const float* __restrict__ s_dst,
                             const float* __restrict__ attn_b,
                             float* __restrict__ score, unsigned* __restrict__ mx_enc) {
    const int e = blockIdx.x * blockDim.x + threadIdx.x;
    if (e >= N_EDGES) return;
    float s = s_src[src[e]] + s_dst[dst[e]] + attn_b[0];
    s = (s >= 0.0f) ? s : NEG_SLOPE * s;
    score[e] = s;
    atomicMax(&mx_enc[dst[e]], fenc(s));
}

// ---- kernel 4: ex = exp(score - mx[dst]); denom[dst] += ex (in-place overwrite) ----
__global__ void edge_exp_k(const int* __restrict__ dst,
                           const unsigned* __restrict__ mx_enc,
                           float* __restrict__ score, float* __restrict__ denom) {
    const int e = blockIdx.x * blockDim.x + threadIdx.x;
    if (e >= N_EDGES) return;
    const int d = dst[e];
    float m = fdec(mx_enc[d]);
    if (!isfinite(m)) m = 0.0f;
    const float ex = expf(score[e] - m);
    score[e] = ex;
    atomicAdd(&denom[d], ex);
}

// ---- kernel 5: h_new[dst] += (ex/denom[dst]) * h[src]; one wave32 per edge ----
// lane k covers dims {k, k+32, k+64, k+96}: coalesced 128B gathers + f32 atomics.
// h / h_new (25.6 MB each) live in the 192MB L2, so the random traffic stays
// on-chip; HBM only streams the edge lists once.
__global__ void edge_aggregate_k(const int* __restrict__ src, const int* __restrict__ dst,
                                 const float* __restrict__ ex, const float* __restrict__ denom,
                                 const float* __restrict__ h, float* __restrict__ out) {
    const int lane = threadIdx.x & 31;
    const int e = (int)(((size_t)blockIdx.x * blockDim.x + threadIdx.x) >> 5);
    if (e >= N_EDGES) return;
    const int s = src[e], d = dst[e];
    const float dn = denom[d];
    const float a = ex[e] / ((dn == 0.0f) ? 1.0f : dn);
    const float* hs = h + (size_t)s * DIM;
    float* od = out + (size_t)d * DIM;
    #pragma unroll
    for (int c = 0; c < 4; ++c) {
        const int k = lane + 32 * c;
        atomicAdd(&od[k], a * hs[k]);
    }
}

extern "C" void kernel_launch(void* const* d_in, const int* in_sizes, int n_in,
                              void* d_out, int out_size, void* d_ws, size_t ws_size,
                              hipStream_t stream) {
    const float* feat    = (const float*)d_in[0];   // [50000,128]
    const int*   src     = (const int*)  d_in[1];   // [1.6M]
    const int*   dst     = (const int*)  d_in[2];   // [1.6M]
    const float* W       = (const float*)d_in[3];   // [128,128]
    const float* bias    = (const float*)d_in[4];   // [128]
    const float* attn_w  = (const float*)d_in[5];   // [256]
    const float* attn_b  = (const float*)d_in[6];   // [1]
    const float* mask_w  = (const float*)d_in[7];   // [128]
    float* out = (float*)d_out;                      // [50000,128]

    // workspace carve-up (bytes)
    char* ws = (char*)d_ws;
    float*    h      = (float*)   (ws);                                  // 25.6 MB
    float*    score  = (float*)   (ws + (size_t)N_NODES * DIM * 4);      //  6.4 MB
    float*    s_src  = (float*)   (ws + 32000000);                       //  0.2 MB
    float*    s_dst  = (float*)   (ws + 32200000);
    unsigned* mx_enc = (unsigned*)(ws + 32400000);
    float*    denom  = (float*)   (ws + 32600000);
    float*    Wp     = (float*)   (ws + 32800000);                       // 64 KB

    // init + weight fold/pack
    init_nodes_k<<<(N_NODES + 255) / 256, 256, 0, stream>>>(mx_enc, denom);
    zero_out_k<<<((size_t)N_NODES * DIM + 255) / 256, 256, 0, stream>>>(out);
    pack_w_k<<<(DIM * DIM + 255) / 256, 256, 0, stream>>>(W, mask_w, Wp);

    // 1: projection GEMM (fp32 WMMA), 3125 row tiles x 8 col-tile waves
    proj_wmma_k<<<dim3(N_NODES / 16), dim3(32, 8), 0, stream>>>(feat, Wp, bias, h);

    // 2: node score GEMVs (wave per node, 8 waves/block)
    node_scores_k<<<(N_NODES * 32 + 255) / 256, 256, 0, stream>>>(h, attn_w, s_src, s_dst);

    // 3-4: edge softmax
    edge_score_k<<<(N_EDGES + 255) / 256, 256, 0, stream>>>(src, dst, s_src, s_dst,
                                                            attn_b, score, mx_enc);
    edge_exp_k<<<(N_EDGES + 255) / 256, 256, 0, stream>>>(dst, mx_enc, score, denom);

    // 5: weighted scatter aggregation (wave per edge)
    edge_aggregate_k<<<((size_t)N_EDGES * 32 + 255) / 256, 256, 0, stream>>>(
        src, dst, score, denom, h, out);
}